// GCN_38431367365260
// MI455X (gfx1250) — compile-verified
//
#include <hip/hip_runtime.h>
#include <hip/hip_bf16.h>

typedef __attribute__((ext_vector_type(2))) float v2f;
typedef __attribute__((ext_vector_type(8))) float v8f;

// ---------------------------------------------------------------------------
// threefry2x32-20 with key = jax.random.key(42) -> (0, 42)
// ---------------------------------------------------------------------------
__device__ __forceinline__ unsigned rotl32(unsigned x, int n) {
    return (x << n) | (x >> (32 - n));
}

__device__ __forceinline__ void threefry2x32_42(unsigned x0, unsigned x1,
                                                unsigned* y0, unsigned* y1) {
    const unsigned k0 = 0u, k1 = 42u;
    const unsigned k2 = 0x1BD11BDAu ^ k0 ^ k1;
    const unsigned ks[3] = {k0, k1, k2};
    const int R[8] = {13, 15, 26, 6, 17, 29, 16, 24};
    x0 += k0; x1 += k1;
#pragma unroll
    for (int g = 0; g < 5; ++g) {
#pragma unroll
        for (int r = 0; r < 4; ++r) {
            x0 += x1;
            x1 = rotl32(x1, R[(g & 1) * 4 + r]);
            x1 ^= x0;
        }
        const int i = g + 1;
        x0 += ks[i % 3];
        x1 += ks[(i + 1) % 3] + (unsigned)i;
    }
    *y0 = x0; *y1 = x1;
}

// keep (scale 2.0) / drop (0.0) matching jax.random.bernoulli(key(42), 0.5, (N,128))
__device__ __forceinline__ float dropout_scale(unsigned i, unsigned half_total) {
    unsigned c0, c1, y0, y1;
    const bool lo = (i < half_total);
    c0 = lo ? i : (i - half_total);
    c1 = lo ? (i + half_total) : i;
    threefry2x32_42(c0, c1, &y0, &y1);
    const unsigned bits = lo ? y0 : y1;
    const float u = __uint_as_float((bits >> 9) | 0x3f800000u) - 1.0f;
    return (u < 0.5f) ? 2.0f : 0.0f;
}

// ---------------------------------------------------------------------------
// zero fill (float4 stores)
// ---------------------------------------------------------------------------
__global__ void zero_kernel(float4* __restrict__ p, long long n4) {
    long long i = (long long)blockIdx.x * blockDim.x + threadIdx.x;
    if (i < n4) p[i] = make_float4(0.f, 0.f, 0.f, 0.f);
}

// ---------------------------------------------------------------------------
// SPMM: out[dst[e]] += edge_weight[e] * x[src[e]]   (F = 128 floats)
// one wave per edge; lane handles 4 consecutive features (float4)
// x (51.2 MB) is expected L2-resident; atomics execute at L2.
// ---------------------------------------------------------------------------
__global__ void spmm_kernel(const float* __restrict__ x,
                            const float* __restrict__ ew,
                            const int* __restrict__ src,
                            const int* __restrict__ dst,
                            float* __restrict__ out, int E) {
    const int wave = (int)((blockIdx.x * (long long)blockDim.x + threadIdx.x) >> 5);
    const int lane = threadIdx.x & 31;
    if (wave >= E) return;
    const int s = src[wave];
    const int d = dst[wave];
    const float w = ew[wave];
    const float4 v = ((const float4*)(x + (size_t)s * 128))[lane];
    float* o = out + (size_t)d * 128 + lane * 4;
    unsafeAtomicAdd(o + 0, v.x * w);   // global_atomic_add_f32
    unsafeAtomicAdd(o + 1, v.y * w);
    unsafeAtomicAdd(o + 2, v.z * w);
    unsafeAtomicAdd(o + 3, v.w * w);
}

// ---------------------------------------------------------------------------
// GEMM via V_WMMA_F32_16X16X4_F32: C[rows,COLS] = A[rows,K] * B[K,COLS] + bias
// One wave computes one 16x16 tile of C, stepping K by 4 (fully unrolled).
// K and COLS are compile-time so ALL strided offsets fold into immediate
// IOFFSETs on the global loads/stores (no per-iteration 64-bit address math).
//
// A frag (16x4 f32, 2 VGPR): lanes 0-15 -> M=lane,   K = k0+{0,1};
//                            lanes16-31 -> M=lane-16, K = k0+{2,3}.
// B frag (4x16 f32, 2 VGPR): lane group selects K pair, lane%16 = N column.
// C/D (8 VGPR): vgpr j -> row j + 8*(lane/16), col = lane%16.
// ---------------------------------------------------------------------------
template <int K, int COLS, bool RELU_DROPOUT>
__global__ void gemm_wmma_kernel(const float* __restrict__ A,
                                 const float* __restrict__ B,
                                 const float* __restrict__ bias,
                                 float* __restrict__ C,
                                 int rows) {
    const int wave = (int)((blockIdx.x * (long long)blockDim.x + threadIdx.x) >> 5);
    const int lane = threadIdx.x & 31;
    constexpr int TILES_N = COLS / 16;
    const int tile_m = wave / TILES_N;
    const int tile_n = wave % TILES_N;
    if (tile_m * 16 >= rows) return;          // wave-uniform: EXEC stays all-1s for WMMA

    const int hsel = lane >> 4;               // 0 or 1
    const int lidx = lane & 15;
    const int arow = tile_m * 16 + lidx;
    const int bcol = tile_n * 16 + lidx;

    // Per-lane bases; every k-step offset below is a compile-time constant.
    const float* __restrict__ ap = A + (unsigned)arow * K + 2 * hsel; // +k0 imm
    const float* __restrict__ bp = B + 2 * hsel * COLS + bcol;        // +k0*COLS imm

    v8f acc = {};
#pragma unroll
    for (int k0 = 0; k0 < K; k0 += 4) {
        v2f av, bv;
        av[0] = ap[k0];
        av[1] = ap[k0 + 1];
        bv[0] = bp[k0 * COLS];
        bv[1] = bp[k0 * COLS + COLS];
        acc = __builtin_amdgcn_wmma_f32_16x16x4_f32(
            /*neg_a=*/false, av, /*neg_b=*/false, bv,
            /*c_mod=*/(short)0, acc, /*reuse_a=*/false, /*reuse_b=*/false);
    }

    const float bval = bias[bcol];
    const int row0 = tile_m * 16 + 8 * hsel;  // rows row0 .. row0+7 in this lane
    float* __restrict__ cp = C + (unsigned)row0 * COLS + bcol;
    const unsigned half_total = (unsigned)(((long long)rows * COLS) >> 1);
#pragma unroll
    for (int j = 0; j < 8; ++j) {
        float v = acc[j] + bval;
        if (RELU_DROPOUT) {
            v = v > 0.f ? v : 0.f;
            v *= dropout_scale((unsigned)(row0 + j) * COLS + (unsigned)bcol,
                               half_total);
        }
        cp[j * COLS] = v;                     // immediate-offset stores
    }
}

// ---------------------------------------------------------------------------
// Pipeline:
//   zero h1 ; spmm(x)->h1 ; gemm1(h1,W1)+relu+dropout->h2 ;
//   zero h1 ; spmm(h2)->h1 ; gemm2(h1,W2)->out
// Workspace: 2 * N * 128 floats (102.4 MB)
// ---------------------------------------------------------------------------
extern "C" void kernel_launch(void* const* d_in, const int* in_sizes, int n_in,
                              void* d_out, int out_size, void* d_ws, size_t ws_size,
                              hipStream_t stream) {
    const float* x  = (const float*)d_in[0];
    const float* ew = (const float*)d_in[1];
    const float* W1 = (const float*)d_in[2];
    const float* b1 = (const float*)d_in[3];
    const float* W2 = (const float*)d_in[4];
    const float* b2 = (const float*)d_in[5];
    const int*   src = (const int*)d_in[6];
    const int*   dst = (const int*)d_in[7];

    const int N = in_sizes[0] / 128;          // 100000
    const int E = in_sizes[1];                // 1600000
    float* out = (float*)d_out;               // [N, 64]

    float* h1 = (float*)d_ws;                 // [N,128] spmm accumulator
    float* h2 = h1 + (size_t)N * 128;         // [N,128] activations

    const long long n4 = (long long)N * 128 / 4;
    const int zgrid = (int)((n4 + 255) / 256);
    const int sgrid = (int)(((long long)E * 32 + 255) / 256);

    // layer 1
    zero_kernel<<<zgrid, 256, 0, stream>>>((float4*)h1, n4);
    spmm_kernel<<<sgrid, 256, 0, stream>>>(x, ew, src, dst, h1, E);
    {
        const int tiles = ((N + 15) / 16) * (128 / 16);
        const int blocks = (tiles + 3) / 4;   // 4 waves (tiles) per 128-thread block
        gemm_wmma_kernel<128, 128, true><<<blocks, 128, 0, stream>>>(h1, W1, b1, h2, N);
    }
    // layer 2
    zero_kernel<<<zgrid, 256, 0, stream>>>((float4*)h1, n4);
    spmm_kernel<<<sgrid, 256, 0, stream>>>(h2, ew, src, dst, h1, E);
    {
        const int tiles = ((N + 15) / 16) * (64 / 16);
        const int blocks = (tiles + 3) / 4;
        gemm_wmma_kernel<128, 64, false><<<blocks, 128, 0, stream>>>(h1, W2, b2, out, N);
    }
}